// STMConv_55551107006919
// MI455X (gfx1250) — compile-verified
//
#include <hip/hip_runtime.h>
#include <hip/hip_bf16.h>

typedef __attribute__((ext_vector_type(2))) float v2f;
typedef __attribute__((ext_vector_type(4))) float v4f;
typedef __attribute__((ext_vector_type(8))) float v8f;

// ---------------------------------------------------------------------------
// Kernel 0: h1[0] = relu(W1[0] @ x + b1[0]).  256->4 1x1 conv; reads x (64MB)
// once from HBM, coalesced across lanes (spatial index is the fast axis).
// ---------------------------------------------------------------------------
__global__ void __launch_bounds__(256) conv1_first_kernel(
    const float* __restrict__ x,    // [16,256,64,64]
    const float* __restrict__ w1,   // [4,256] (block 0)
    const float* __restrict__ b1,   // [4]     (block 0)
    float* __restrict__ h1out)      // [16,4,64,64]
{
    int p  = blockIdx.x * 256 + threadIdx.x;    // 0 .. 65535 pixels
    int n  = p >> 12;                            // image
    int sp = p & 4095;                           // y*64+x
    const float* xp = x + (size_t)n * 256 * 4096 + sp;

    float a0 = b1[0], a1 = b1[1], a2 = b1[2], a3 = b1[3];
#pragma unroll 8
    for (int k = 0; k < 256; ++k) {
        float xv = xp[(size_t)k * 4096];         // coalesced across lanes
        a0 = fmaf(w1[0 * 256 + k], xv, a0);      // uniform (scalarized) weights
        a1 = fmaf(w1[1 * 256 + k], xv, a1);
        a2 = fmaf(w1[2 * 256 + k], xv, a2);
        a3 = fmaf(w1[3 * 256 + k], xv, a3);
    }
    float* o = h1out + (size_t)n * 4 * 4096 + sp;
    o[0]        = fmaxf(a0, 0.f);
    o[4096]     = fmaxf(a1, 0.f);
    o[2 * 4096] = fmaxf(a2, 0.f);
    o[3 * 4096] = fmaxf(a3, 0.f);
}

// ---------------------------------------------------------------------------
// Fused per-block kernel (32 launches; kernel boundary = grid sync for halo):
//   h2  = relu(conv3x3(h1[i]) + b2)                        (VALU, 4ch)
//   h3  = relu(W3 @ h2 + b3)           via v_wmma_f32_16x16x4_f32 (K=4)
//   acc (+)= h3                        (batched L2-resident rmw)
//   h1[i+1] = relu(W1' @ h3 + b1')     via 64 chained K=4 WMMAs
// One WG = one 16x16 pixel tile of one image; 256 threads = 8 waves,
// wave w owns pixel rows g=2w,2w+1 (16 px each = WMMA N dimension).
// FIRST: acc store instead of rmw.  HAS_NEXT: produce h1[i+1].
// ---------------------------------------------------------------------------
template <bool FIRST, bool HAS_NEXT>
__global__ void __launch_bounds__(256) fused_block_kernel(
    const float* __restrict__ h1in,  // [16,4,64,64]
    float* __restrict__ h1out,       // [16,4,64,64]
    float* __restrict__ acc,         // [16,256,64,64] == d_out
    const float* __restrict__ w2,    // [4,4,3,3]
    const float* __restrict__ b2,    // [4]
    const float* __restrict__ w3,    // [256,4]
    const float* __restrict__ b3,    // [256]
    const float* __restrict__ w1n,   // [4,256] next block
    const float* __restrict__ b1n)   // [4]     next block
{
    __shared__ float h1s[4][18][20];              // h1 tile + 1px halo
    __shared__ float h2s[4 * 272];                // [4ch][256px], padded
    __shared__ __align__(16) float w3s[1024];     // [256][4]
    __shared__ __align__(16) float w1s[16 * 260]; // [16][256] rows 4..15 = 0
    __shared__ float w2s[144];
    __shared__ float b2s[4];
    __shared__ float b1s[4];
    __shared__ __align__(16) float b3s[256];
    __shared__ __align__(16) float h3s[8][16 * 20]; // per-wave [16px][16ch]

    const int tid = threadIdx.x;
    const int bid = blockIdx.x;                   // 16 images * 16 tiles
    const int n  = bid >> 4;
    const int tt = bid & 15;
    const int y0 = (tt >> 2) * 16;
    const int x0 = (tt & 3) * 16;

    // ---- stage weights ----
    for (int e = tid; e < 1024; e += 256) w3s[e] = w3[e];
    if (HAS_NEXT) {
        // single race-free pass: rows 0..3 real, rest (incl. padding) zero
        for (int e = tid; e < 16 * 260; e += 256) {
            int r = e / 260, col = e - r * 260;
            float v = 0.f;
            if (r < 4 && col < 256) v = w1n[r * 256 + col];
            w1s[e] = v;
        }
    }
    if (tid < 144) w2s[tid] = w2[tid];
    if (tid < 4) { b2s[tid] = b2[tid]; if (HAS_NEXT) b1s[tid] = b1n[tid]; }
    b3s[tid] = b3[tid];

    // ---- stage h1 tile + halo (SAME padding -> zero fill) ----
    for (int e = tid; e < 4 * 18 * 18; e += 256) {
        int c   = e / 324;
        int r   = (e % 324) / 18;
        int col = e % 18;
        int gy = y0 + r - 1, gx = x0 + col - 1;
        float v = 0.f;
        if (gy >= 0 && gy < 64 && gx >= 0 && gx < 64)
            v = h1in[((size_t)(n * 4 + c)) * 4096 + gy * 64 + gx];
        h1s[c][r][col] = v;
    }
    __syncthreads();

    // ---- conv2: 3x3, 4->4 ch, one pixel per thread (144 FMA) ----
    {
        const int ly = tid >> 4, lx = tid & 15;
#pragma unroll
        for (int c = 0; c < 4; ++c) {
            float s = b2s[c];
#pragma unroll
            for (int k = 0; k < 4; ++k)
#pragma unroll
                for (int dy = 0; dy < 3; ++dy)
#pragma unroll
                    for (int dx = 0; dx < 3; ++dx)
                        s = fmaf(w2s[(c * 4 + k) * 9 + dy * 3 + dx],
                                 h1s[k][ly + dy][lx + dx], s);
            h2s[c * 272 + tid] = fmaxf(s, 0.f);
        }
    }
    // wave's pixel rows were produced by its own threads: intra-wave LDS RAW
    // handled by dscnt waits, no WG barrier needed.

    const int wv   = tid >> 5;
    const int lane = tid & 31;
    const int half = lane >> 4;           // lanes 16-31 hold K=2,3 / M+8
    const int m    = lane & 15;
    float* h3c = &h3s[wv][0];             // [16px][16ch], stride 20

    // ---- preload all conv3 A operands (W3 chunks) into registers ----
    // (constant-indexed below thanks to full unroll -> no v_movrels)
    v2f aW3r[16];
#pragma unroll
    for (int c = 0; c < 16; ++c)
        aW3r[c] = *(const v2f*)&w3s[(16 * c + m) * 4 + 2 * half];

    for (int g = 2 * wv; g < 2 * wv + 2; ++g) {
        const int gy  = y0 + g;
        const int sp  = gy * 64 + x0 + m;
        float* accg = acc + (size_t)n * 256 * 4096 + sp;

        // B operand for conv3: h2 [4 x 16px]
        v2f bH2;
        bH2.x = h2s[(half ? 2 : 0) * 272 + g * 16 + m];
        bH2.y = h2s[(half ? 3 : 1) * 272 + g * 16 + m];

        v8f cD = {};                      // conv1-next accumulator

#pragma unroll
        for (int c = 0; c < 16; ++c) {    // FULL unroll: constant reg indices
            v8f d = {};
            d = __builtin_amdgcn_wmma_f32_16x16x4_f32(
                    false, aW3r[c], false, bH2, (short)0, d, false, false);

            // bias + relu (vectorized LDS bias reads)
            v4f blo = *(const v4f*)&b3s[16 * c + 8 * half];
            v4f bhi = *(const v4f*)&b3s[16 * c + 8 * half + 4];
            v4f lo, hi;
#pragma unroll
            for (int j = 0; j < 4; ++j) lo[j] = fmaxf(d[j]     + blo[j], 0.f);
#pragma unroll
            for (int j = 0; j < 4; ++j) hi[j] = fmaxf(d[j + 4] + bhi[j], 0.f);

            // LDS transpose stage: [px][ch], 2x ds_store_b128
            *(v4f*)&h3c[m * 20 + 8 * half]     = lo;
            *(v4f*)&h3c[m * 20 + 8 * half + 4] = hi;

            // batched global acc update (8 independent loads, then stores)
            float* accp = accg + ((size_t)(16 * c + 8 * half)) * 4096;
            if (FIRST) {
#pragma unroll
                for (int j = 0; j < 4; ++j) accp[(size_t)j * 4096]       = lo[j];
#pragma unroll
                for (int j = 0; j < 4; ++j) accp[(size_t)(j + 4) * 4096] = hi[j];
            } else {
                float av[8];
#pragma unroll
                for (int j = 0; j < 8; ++j) av[j] = accp[(size_t)j * 4096];
#pragma unroll
                for (int j = 0; j < 4; ++j) accp[(size_t)j * 4096]       = av[j]     + lo[j];
#pragma unroll
                for (int j = 0; j < 4; ++j) accp[(size_t)(j + 4) * 4096] = av[j + 4] + hi[j];
            }

            if (HAS_NEXT) {
                // 4 K-steps consume exactly this 16-channel chunk
#pragma unroll
                for (int t = 0; t < 4; ++t) {
                    v2f aW1 = *(const v2f*)&w1s[m * 260 + 16 * c + 4 * t + 2 * half];
                    v2f bH3 = *(const v2f*)&h3c[m * 20 + 4 * t + 2 * half];
                    cD = __builtin_amdgcn_wmma_f32_16x16x4_f32(
                            false, aW1, false, bH3, (short)0, cD, false, false);
                }
            }
        }

        if (HAS_NEXT && half == 0) {      // D rows 0..3 live in lanes 0-15
#pragma unroll
            for (int j = 0; j < 4; ++j) {
                float v = fmaxf(cD[j] + b1s[j], 0.f);
                h1out[((size_t)(n * 4 + j)) * 4096 + sp] = v;
            }
        }
    }
}

// ---------------------------------------------------------------------------
// Host-side orchestration: 1 + 32 launches, all on `stream`, graph-capturable.
// h1 ping-pong buffers live in d_ws (2 MB); d_out is the accumulator.
// ---------------------------------------------------------------------------
extern "C" void kernel_launch(void* const* d_in, const int* in_sizes, int n_in,
                              void* d_out, int out_size, void* d_ws, size_t ws_size,
                              hipStream_t stream) {
    const float* x  = (const float*)d_in[0];
    const float* w1 = (const float*)d_in[1];   // [32,4,256]
    const float* b1 = (const float*)d_in[2];   // [32,4]
    const float* w2 = (const float*)d_in[3];   // [32,4,4,3,3]
    const float* b2 = (const float*)d_in[4];   // [32,4]
    const float* w3 = (const float*)d_in[5];   // [32,256,4]
    const float* b3 = (const float*)d_in[6];   // [32,256]
    float* acc = (float*)d_out;

    const size_t H1N = (size_t)16 * 4 * 64 * 64;   // 262144 floats = 1MB
    float* buf0 = (float*)d_ws;
    float* buf1 = buf0 + H1N;

    conv1_first_kernel<<<256, 256, 0, stream>>>(x, w1, b1, buf0);

    for (int i = 0; i < 32; ++i) {
        const float* h1i = (i & 1) ? buf1 : buf0;
        float*       h1o = (i & 1) ? buf0 : buf1;
        const float* w1next = (i < 31) ? (w1 + (size_t)(i + 1) * 1024) : nullptr;
        const float* b1next = (i < 31) ? (b1 + (size_t)(i + 1) * 4)    : nullptr;
        const float* w2i = w2 + (size_t)i * 144;
        const float* b2i = b2 + (size_t)i * 4;
        const float* w3i = w3 + (size_t)i * 1024;
        const float* b3i = b3 + (size_t)i * 256;
        if (i == 0) {
            fused_block_kernel<true, true><<<256, 256, 0, stream>>>(
                h1i, h1o, acc, w2i, b2i, w3i, b3i, w1next, b1next);
        } else if (i < 31) {
            fused_block_kernel<false, true><<<256, 256, 0, stream>>>(
                h1i, h1o, acc, w2i, b2i, w3i, b3i, w1next, b1next);
        } else {
            fused_block_kernel<false, false><<<256, 256, 0, stream>>>(
                h1i, h1o, acc, w2i, b2i, w3i, b3i, w1next, b1next);
        }
    }
}